// SparseMatrix_19078244729236
// MI455X (gfx1250) — compile-verified
//
#include <hip/hip_runtime.h>

typedef __attribute__((ext_vector_type(2))) float v2f;
typedef __attribute__((ext_vector_type(8))) float v8f;

#define KDIM 128

// ---------------------------------------------------------------------------
// Kernel 1: zero-init the output accumulator (harness poisons d_out).
// ---------------------------------------------------------------------------
__global__ void spmm_zero_kernel(float* __restrict__ out, long long n) {
  long long n4 = n >> 2;
  long long i = (long long)blockIdx.x * blockDim.x + threadIdx.x;
  long long stride = (long long)gridDim.x * blockDim.x;
  float4* out4 = (float4*)out;
  float4 z = make_float4(0.f, 0.f, 0.f, 0.f);
  for (long long j = i; j < n4; j += stride) out4[j] = z;
  // scalar tail (out_size is divisible by 4 here, but stay safe)
  long long tail = n & 3LL;
  if (i < tail) out[n4 * 4 + i] = 0.f;
}

// ---------------------------------------------------------------------------
// Kernel 2: COO SpMM via V_WMMA_F32_16X16X4_F32.
// One wave processes a chunk of 16 nonzeros:
//   G[m][n] = v_m * y[col_m][kt*16 + n]   (16x16 tile, 8 tiles cover K=128)
// computed as sum of four 16x16x4 WMMAs with A_j = diagonal slice of diag(v),
// then scattered with one global_atomic_add_f32 per element.
// ---------------------------------------------------------------------------
__global__ __launch_bounds__(256) void spmm_wmma_kernel(
    const int* __restrict__ rowIdx, const int* __restrict__ colIdx,
    const float* __restrict__ vals, const float* __restrict__ y,
    float* __restrict__ out, int nnz) {
  const int lane = threadIdx.x & 31;
  const int m  = lane & 15;   // row-within-tile (A), column N (B/C/D)
  const int hi = lane >> 4;   // half-wave selector

  const long long waveId =
      (long long)blockIdx.x * (blockDim.x >> 5) + (threadIdx.x >> 5);
  const long long nWaves = (long long)gridDim.x * (blockDim.x >> 5);

  for (long long base = waveId * 16; base < (long long)nnz;
       base += nWaves * 16) {
    // --- load chunk metadata: lane L fetches nonzero (base + L%16) ---------
    long long i = base + m;
    bool valid = i < (long long)nnz;
    int   rw = valid ? rowIdx[i] : 0;
    int   cl = valid ? colIdx[i] : 0;
    float vm = valid ? vals[i]   : 0.0f;  // lane L holds v_{L%16} already

    // --- prefetch next chunk's streams (global_prefetch_b8) ----------------
    long long nb = base + nWaves * 16;
    if (nb + m < (long long)nnz) {
      __builtin_prefetch(rowIdx + nb + m, 0, 1);
      __builtin_prefetch(colIdx + nb + m, 0, 1);
      __builtin_prefetch(vals   + nb + m, 0, 1);
    }

    // --- build A_j (16x4 f32): lane L, vgpr p holds A[m][k], k = p + 2*hi --
    // A_j[m][k] = v_m if (m == 4*j + k) else 0  -> sum_j A_j*B_j = diag(v)*Y
    v2f a[4];
    const int k0 = 2 * hi;
#pragma unroll
    for (int j = 0; j < 4; ++j) {
      a[j].x = (m == 4 * j + k0)     ? vm : 0.0f;
      a[j].y = (m == 4 * j + k0 + 1) ? vm : 0.0f;
    }

    // --- scatter rows: C/D vgpr p, half hi <-> matrix row (p + 8*hi) -------
    int rp[8];
#pragma unroll
    for (int p = 0; p < 8; ++p) rp[p] = __shfl(rw, p + 8 * hi, 32);

    // --- gather pointers for B_j: vgpr p, half hi reads y[col_{4j+2hi+p}] --
    const float* yb[8];
#pragma unroll
    for (int j = 0; j < 4; ++j) {
      int c0 = __shfl(cl, 4 * j + 2 * hi + 0, 32);
      int c1 = __shfl(cl, 4 * j + 2 * hi + 1, 32);
      yb[2 * j + 0] = y + (size_t)c0 * KDIM;
      yb[2 * j + 1] = y + (size_t)c1 * KDIM;
    }

    // --- 8 K-tiles of 16 columns each --------------------------------------
#pragma unroll 2
    for (int kt = 0; kt < 8; ++kt) {
      const int co = kt * 16 + m;  // column within y/out row
      v8f acc = {};
#pragma unroll
      for (int j = 0; j < 4; ++j) {
        v2f b;
        b.x = yb[2 * j + 0][co];
        b.y = yb[2 * j + 1][co];
        // D = A_j x B_j + C   (16x16x4 f32 WMMA)
        acc = __builtin_amdgcn_wmma_f32_16x16x4_f32(
            false, a[j], false, b, (short)0, acc, false, false);
      }
      // --- scatter-add: one atomic per produced element ---------------------
#pragma unroll
      for (int p = 0; p < 8; ++p) {
        atomicAdd(out + (size_t)rp[p] * KDIM + co, acc[p]);
      }
    }
  }
}

// ---------------------------------------------------------------------------
// Launch
// ---------------------------------------------------------------------------
extern "C" void kernel_launch(void* const* d_in, const int* in_sizes, int n_in,
                              void* d_out, int out_size, void* d_ws,
                              size_t ws_size, hipStream_t stream) {
  const int*   index = (const int*)d_in[0];   // [2, nnz] COO indices
  const float* vals  = (const float*)d_in[1]; // [nnz]
  const float* y     = (const float*)d_in[2]; // [n, K] dense
  float*       out   = (float*)d_out;         // [m, K] dense

  const int nnz = in_sizes[1];
  const int* rowIdx = index;
  const int* colIdx = index + nnz;

  // zero the accumulator
  spmm_zero_kernel<<<4096, 256, 0, stream>>>(out, (long long)out_size);

  // one wave per 16-nnz chunk, grid-strided
  const int wavesPerBlock = 256 / 32;
  long long chunks = ((long long)nnz + 15) / 16;
  long long blocksNeeded = (chunks + wavesPerBlock - 1) / wavesPerBlock;
  int blocks = (blocksNeeded < 16384LL) ? (int)blocksNeeded : 16384;
  if (blocks < 1) blocks = 1;

  spmm_wmma_kernel<<<blocks, 256, 0, stream>>>(rowIdx, colIdx, vals, y, out,
                                               nnz);
}